// GCN_40827959116202
// MI455X (gfx1250) — compile-verified
//
#include <hip/hip_runtime.h>

typedef __attribute__((ext_vector_type(2))) float v2f;
typedef __attribute__((ext_vector_type(8))) float v8f;

#define N_NODES 32768
#define CIN 128
#define CHID 256
#define COUTC 128
#define N_EDGES 1048576

// ---------------- degree / norm ----------------
__global__ void deg_init_kernel(float* __restrict__ deg) {
    int i = blockIdx.x * blockDim.x + threadIdx.x;
    if (i < N_NODES) deg[i] = 1.0f;  // self-loop contributes 1
}

__global__ void deg_count_kernel(const int* __restrict__ dst, float* __restrict__ deg) {
    int i = blockIdx.x * blockDim.x + threadIdx.x;
    if (i < N_EDGES) atomicAdd(&deg[dst[i]], 1.0f);
}

__global__ void dinv_kernel(float* __restrict__ deg) {
    int i = blockIdx.x * blockDim.x + threadIdx.x;
    if (i < N_NODES) {
        float d = deg[i];
        deg[i] = (d > 0.0f) ? rsqrtf(d) : 0.0f;
    }
}

// ---------------- f32 WMMA GEMM: Out[NR x COUT] = A[NR x K] * W[K x COUT] ----
// One wave computes a 16x64 strip using V_WMMA_F32_16X16X4_F32.
template <int K, int COUT>
__global__ void gemm_wmma_kernel(const float* __restrict__ A,
                                 const float* __restrict__ W,
                                 float* __restrict__ Out) {
    constexpr int TILES_N = COUT / 64;
    const int wave  = blockIdx.x * (blockDim.x >> 5) + (threadIdx.x >> 5);
    const int lane  = threadIdx.x & 31;
    const int tm    = wave / TILES_N;          // 16-row tile index
    const int tn    = (wave % TILES_N) * 64;   // 64-col strip start
    const int col   = lane & 15;               // M index for A frag, N index for B/C frag
    const int khalf = lane >> 4;               // 0: K={0,1}  1: K={2,3}

    v8f acc0 = {}, acc1 = {}, acc2 = {}, acc3 = {};
    const float* Ap = A + (size_t)(tm * 16 + col) * K;
    const float* Wp = W + tn + col;

#pragma unroll 4
    for (int k = 0; k < K; k += 4) {
        const int kk = k + 2 * khalf;
        v2f a;
        a.x = Ap[kk];
        a.y = Ap[kk + 1];
        const float* w0 = Wp + (size_t)kk * COUT;
        v2f b0, b1, b2, b3;
        b0.x = w0[0];   b0.y = w0[COUT + 0];
        b1.x = w0[16];  b1.y = w0[COUT + 16];
        b2.x = w0[32];  b2.y = w0[COUT + 32];
        b3.x = w0[48];  b3.y = w0[COUT + 48];
        acc0 = __builtin_amdgcn_wmma_f32_16x16x4_f32(false, a, false, b0, (short)0, acc0, false, false);
        acc1 = __builtin_amdgcn_wmma_f32_16x16x4_f32(false, a, false, b1, (short)0, acc1, false, false);
        acc2 = __builtin_amdgcn_wmma_f32_16x16x4_f32(false, a, false, b2, (short)0, acc2, false, false);
        acc3 = __builtin_amdgcn_wmma_f32_16x16x4_f32(false, a, false, b3, (short)0, acc3, false, false);
    }

    // C/D layout: VGPR v, lanes 0-15 -> M=v, lanes 16-31 -> M=v+8; N = lane&15
    float* Op = Out + (size_t)(tm * 16 + 8 * khalf) * COUT + tn + col;
#pragma unroll
    for (int v = 0; v < 8; ++v) {
        Op[(size_t)v * COUT + 0]  = acc0[v];
        Op[(size_t)v * COUT + 16] = acc1[v];
        Op[(size_t)v * COUT + 32] = acc2[v];
        Op[(size_t)v * COUT + 48] = acc3[v];
    }
}

// ---------------- edge aggregation: out[dst] += h[src] * dinv[src]*dinv[dst] --
// One wave per edge (grid-strided); lane covers C/32 contiguous channels.
template <int C>
__global__ void aggregate_kernel(const float* __restrict__ h,
                                 const int* __restrict__ src,
                                 const int* __restrict__ dst,
                                 const float* __restrict__ dinv,
                                 float* __restrict__ out) {
    constexpr int PER = C / 32;
    const int wavesPerBlock = blockDim.x >> 5;
    const int lane = threadIdx.x & 31;
    const long long total  = (long long)N_EDGES + N_NODES;
    const long long stride = (long long)gridDim.x * wavesPerBlock;
    for (long long e = (long long)blockIdx.x * wavesPerBlock + (threadIdx.x >> 5);
         e < total; e += stride) {
        int s, d;
        if (e < (long long)N_EDGES) { s = src[e]; d = dst[e]; }
        else                        { s = d = (int)(e - N_EDGES); }
        const float w = dinv[s] * dinv[d];
        const float* hp = h   + (size_t)s * C + lane * PER;
        float*       op = out + (size_t)d * C + lane * PER;
#pragma unroll
        for (int j = 0; j < PER; j += 4) {
            float4 v = *reinterpret_cast<const float4*>(hp + j);
            atomicAdd(op + j + 0, v.x * w);
            atomicAdd(op + j + 1, v.y * w);
            atomicAdd(op + j + 2, v.z * w);
            atomicAdd(op + j + 3, v.w * w);
        }
    }
}

// ---------------- BatchNorm ----------------
__global__ void bn_stats_kernel(const float* __restrict__ agg,
                                const float* __restrict__ b1,
                                float* __restrict__ stats,  // [sum(256), sumsq(256)]
                                int rowsPerBlock) {
    const int c = threadIdx.x;          // blockDim.x == CHID, coalesced across channels
    const float bias = b1[c];
    const int r0 = blockIdx.x * rowsPerBlock;
    float s = 0.0f, s2 = 0.0f;
    for (int r = 0; r < rowsPerBlock; ++r) {
        float v = agg[(size_t)(r0 + r) * CHID + c] + bias;
        s  += v;
        s2 += v * v;
    }
    atomicAdd(&stats[c],        s);
    atomicAdd(&stats[CHID + c], s2);
}

__global__ void bn_params_kernel(const float* __restrict__ stats,
                                 const float* __restrict__ gamma,
                                 const float* __restrict__ beta,
                                 float* __restrict__ params) {  // [scale(256), shift(256)]
    const int c = threadIdx.x;
    const float invN = 1.0f / (float)N_NODES;
    const float mean = stats[c] * invN;
    const float var  = stats[CHID + c] * invN - mean * mean;
    const float sc   = gamma[c] * rsqrtf(var + 1e-5f);
    params[c]        = sc;
    params[CHID + c] = beta[c] - mean * sc;
}

__global__ void bn_relu_kernel(const float* __restrict__ agg,
                               const float* __restrict__ b1,
                               const float* __restrict__ params,
                               float* __restrict__ outh) {
    const size_t i = (size_t)blockIdx.x * blockDim.x + threadIdx.x;
    const int c = (int)(i & (CHID - 1));
    float v = (agg[i] + b1[c]) * params[c] + params[CHID + c];
    outh[i] = (v > 0.0f) ? v : 0.0f;
}

__global__ void out_init_kernel(float* __restrict__ out, const float* __restrict__ b2) {
    const size_t i = (size_t)blockIdx.x * blockDim.x + threadIdx.x;
    out[i] = b2[i & (COUTC - 1)];
}

// ---------------- host-side orchestration ----------------
extern "C" void kernel_launch(void* const* d_in, const int* in_sizes, int n_in,
                              void* d_out, int out_size, void* d_ws, size_t ws_size,
                              hipStream_t stream) {
    const float* x     = (const float*)d_in[0];
    const int*   ei    = (const int*)d_in[1];
    const float* W1    = (const float*)d_in[2];
    const float* b1    = (const float*)d_in[3];
    const float* gamma = (const float*)d_in[4];
    const float* beta  = (const float*)d_in[5];
    const float* W2    = (const float*)d_in[6];
    const float* b2    = (const float*)d_in[7];
    float*       out   = (float*)d_out;
    const int* src = ei;             // edge_index[0]
    const int* dst = ei + N_EDGES;   // edge_index[1]

    float* ws     = (float*)d_ws;
    float* dinv   = ws;                                   // N
    float* buf1   = ws + N_NODES;                         // N*CHID : h1, later h_bn
    float* buf2   = buf1 + (size_t)N_NODES * CHID;        // N*CHID : agg1, later h2
    float* stats  = buf2 + (size_t)N_NODES * CHID;        // 512
    float* params = stats + 2 * CHID;                     // 512

    hipMemsetAsync(stats, 0, 2 * CHID * sizeof(float), stream);
    hipMemsetAsync(buf2, 0, (size_t)N_NODES * CHID * sizeof(float), stream);

    deg_init_kernel<<<N_NODES / 256, 256, 0, stream>>>(dinv);
    deg_count_kernel<<<N_EDGES / 256, 256, 0, stream>>>(dst, dinv);
    dinv_kernel<<<N_NODES / 256, 256, 0, stream>>>(dinv);

    // conv1 linear: buf1 = x @ W1   (32768x128 * 128x256)
    gemm_wmma_kernel<CIN, CHID>
        <<<(N_NODES / 16) * (CHID / 64) / 8, 256, 0, stream>>>(x, W1, buf1);

    // conv1 aggregation: buf2 += buf1[src]*norm
    aggregate_kernel<CHID><<<8192, 256, 0, stream>>>(buf1, src, dst, dinv, buf2);

    // BN statistics + params + apply(ReLU) -> buf1
    bn_stats_kernel<<<256, CHID, 0, stream>>>(buf2, b1, stats, N_NODES / 256);
    bn_params_kernel<<<1, CHID, 0, stream>>>(stats, gamma, beta, params);
    bn_relu_kernel<<<(N_NODES * CHID) / 256, 256, 0, stream>>>(buf2, b1, params, buf1);

    // conv2 linear: buf2 = buf1 @ W2   (32768x256 * 256x128)
    gemm_wmma_kernel<CHID, COUTC>
        <<<(N_NODES / 16) * (COUTC / 64) / 8, 256, 0, stream>>>(buf1, W2, buf2);

    // output = b2 broadcast, then aggregate h2
    out_init_kernel<<<(N_NODES * COUTC) / 256, 256, 0, stream>>>(out, b2);
    aggregate_kernel<COUTC><<<8192, 256, 0, stream>>>(buf2, src, dst, dinv, out);
}